// TripletLoss_19516331393602
// MI455X (gfx1250) — compile-verified
//
#include <hip/hip_runtime.h>
#include <hip/hip_bf16.h>

typedef float v2f __attribute__((ext_vector_type(2)));
typedef float v4f __attribute__((ext_vector_type(4)));
typedef float v8f __attribute__((ext_vector_type(8)));

#define D_DIM 128
#define NJC 20   // j-chunks per i-tile

// ---------------------------------------------------------------------------
// Kernel 1: row squared-norms sq[n] straight from the two source arrays.
// One wave per row; each lane handles a float4 (32*4 = 128).
// ---------------------------------------------------------------------------
__global__ __launch_bounds__(256) void sq_kernel(
    const float* __restrict__ inputs, const float* __restrict__ mean_neg,
    float* __restrict__ sq, int B, int n) {
  int row = blockIdx.x * 8 + (threadIdx.x >> 5);
  int lane = threadIdx.x & 31;
  if (row >= n) return;
  const float* src = (row < B) ? (inputs + (size_t)row * D_DIM)
                               : (mean_neg + (size_t)(row - B) * D_DIM);
  float4 v = ((const float4*)src)[lane];
  float s = v.x * v.x + v.y * v.y + v.z * v.z + v.w * v.w;
  #pragma unroll
  for (int m = 16; m >= 1; m >>= 1) s += __shfl_xor(s, m, 32);
  if (lane == 0) sq[row] = s;
}

// ---------------------------------------------------------------------------
// Kernel 2: concat labels, init per-row max(ap)=0 / min(an)=1e30 (as uints;
// all distances are >= 0 so uint compare order == float order).
// ---------------------------------------------------------------------------
__global__ void prep_labels_kernel(const int* __restrict__ tgt,
                                   const int* __restrict__ mlab,
                                   int* __restrict__ lab,
                                   unsigned* __restrict__ apb,
                                   unsigned* __restrict__ anb, int B, int n) {
  int i = blockIdx.x * 256 + threadIdx.x;
  if (i >= n) return;
  lab[i] = (i < B) ? tgt[i] : mlab[i - B];
  apb[i] = 0u;                       // 0.0f
  anb[i] = __float_as_uint(1e30f);   // +1e30
}

// ---------------------------------------------------------------------------
// Kernel 3: swizzle features into the WMMA 16x4-f32 lane layout.
// featsW[t][kp][lane] (float4): for lane l, rows t*16 + l%16, K offsets
// kp*8 + (l>=16?2:0) + {0,1} (elements .xy -> k=2kp) and +4 (.zw -> k=2kp+1).
// A-tiles and B-tiles both read this array with fully coalesced b128 loads.
// Wave per (tile t, kpair kp); pays the divergent row-gather exactly once.
// ---------------------------------------------------------------------------
__global__ __launch_bounds__(256) void swizzle_kernel(
    const float* __restrict__ inputs, const float* __restrict__ mean_neg,
    float* __restrict__ featsW, int B, int ntiles) {
  int gw = blockIdx.x * 8 + (threadIdx.x >> 5);
  int lane = threadIdx.x & 31;
  int t = gw >> 4;          // tile index
  int kp = gw & 15;         // k-pair index (covers k = 2kp, 2kp+1)
  if (t >= ntiles) return;
  int srow = t * 16 + (lane & 15);
  int koff = (lane >> 4) << 1;
  const float* src = (srow < B) ? (inputs + (size_t)srow * D_DIM)
                                : (mean_neg + (size_t)(srow - B) * D_DIM);
  v2f lo = *(const v2f*)(src + kp * 8 + koff);       // k = 2kp
  v2f hi = *(const v2f*)(src + kp * 8 + 4 + koff);   // k = 2kp+1
  v4f out;
  out.x = lo.x; out.y = lo.y; out.z = hi.x; out.w = hi.y;
  ((v4f*)featsW)[((size_t)(t * 16 + kp) << 5) + lane] = out;
}

// ---------------------------------------------------------------------------
// Kernel 4: Gram tiles via f32 WMMA + fused distance / hard-example mining.
// Wave w: i-tile = w / NJC, j-chunk = w % NJC. 8 waves per block.
// All hot-loop loads use a base pointer advanced once per tile + immediate
// offsets, so the loop body is just clause loads + WMMAs + the epilogue.
// ---------------------------------------------------------------------------
__global__ __launch_bounds__(256) void gram_mine_kernel(
    const float* __restrict__ featsW, const float* __restrict__ sq,
    const int* __restrict__ lab, unsigned* __restrict__ apb,
    unsigned* __restrict__ anb, int ntiles, int jchunk) {
  const int lane = threadIdx.x & 31;
  const int gw = blockIdx.x * 8 + (threadIdx.x >> 5);
  const int itile = gw / NJC;
  const int jc = gw % NJC;
  if (itile >= ntiles) return;
  const int i0 = itile * 16;
  const v4f* fw4 = (const v4f*)featsW;

  // Preload A strip: 16 coalesced b128 loads (immediate offsets kp*512B).
  v2f a[32];
  {
    const v4f* pa = fw4 + (((size_t)itile * 16) << 5) + lane;
    #pragma unroll
    for (int kp = 0; kp < 16; ++kp) {
      v4f av = pa[kp << 5];
      a[2 * kp] = av.lo;
      a[2 * kp + 1] = av.hi;
    }
  }

  // Per-row constants for this lane's 8 C rows: M = r + (lane>=16 ? 8 : 0).
  float sqi[8];
  int labi[8];
  const int rbase = i0 + ((lane >> 4) << 3);
  #pragma unroll
  for (int r = 0; r < 8; ++r) {
    sqi[r] = sq[rbase + r];
    labi[r] = lab[rbase + r];
  }

  float ap[8], an[8];
  #pragma unroll
  for (int r = 0; r < 8; ++r) { ap[r] = 0.0f; an[r] = 1e30f; }

  int jt_beg = jc * jchunk;
  int jt_end = jt_beg + jchunk;
  if (jt_end > ntiles) jt_end = ntiles;

  // Per-tile walking pointers: one 64-bit add each per iteration.
  const v4f* pb = fw4 + (((size_t)jt_beg * 16) << 5) + lane;
  const float* psq = sq + jt_beg * 16 + (lane & 15);
  const int* plab = lab + jt_beg * 16 + (lane & 15);

  for (int jt = jt_beg; jt < jt_end; ++jt) {
    const float sqj = *psq;
    const int labj = *plab;
    psq += 16;
    plab += 16;

    // Stage the B tile: 16 immediate-offset b128 loads into distinct regs.
    v4f b[16];
    #pragma unroll
    for (int kp = 0; kp < 16; ++kp) b[kp] = pb[kp << 5];
    pb += 512;  // next tile (16 kp * 32 lanes)

    v8f c = {};
    #pragma unroll
    for (int kp = 0; kp < 16; ++kp) {
      c = __builtin_amdgcn_wmma_f32_16x16x4_f32(
          false, a[2 * kp], false, b[kp].lo, (short)0, c, false, false);
      c = __builtin_amdgcn_wmma_f32_16x16x4_f32(
          false, a[2 * kp + 1], false, b[kp].hi, (short)0, c, false, false);
    }

    #pragma unroll
    for (int r = 0; r < 8; ++r) {
      float d2 = __builtin_fmaf(-2.0f, c[r], sqi[r] + sqj);
      float d = __builtin_sqrtf(fmaxf(d2, 1e-12f));
      if (labi[r] == labj) {
        ap[r] = fmaxf(ap[r], d);
      } else {
        an[r] = fminf(an[r], d);
      }
    }
  }

  // Reduce across the 16 lanes of each half (xor masks < 16 stay in-group),
  // then lanes 0 and 16 publish their 8 rows.
  #pragma unroll
  for (int r = 0; r < 8; ++r) {
    float av = ap[r], nv = an[r];
    #pragma unroll
    for (int m = 8; m >= 1; m >>= 1) {
      av = fmaxf(av, __shfl_xor(av, m, 32));
      nv = fminf(nv, __shfl_xor(nv, m, 32));
    }
    if ((lane & 15) == 0) {
      atomicMax(&apb[rbase + r], __float_as_uint(av));
      atomicMin(&anb[rbase + r], __float_as_uint(nv));
    }
  }
}

// ---------------------------------------------------------------------------
// Kernel 5: final scalar reduction -> (loss, prec, dist_p, dist_n).
// ---------------------------------------------------------------------------
__global__ __launch_bounds__(256) void finalize_kernel(
    const unsigned* __restrict__ apb, const unsigned* __restrict__ anb,
    float* __restrict__ out, int n) {
  __shared__ float sm[4][256];
  float loss = 0.0f, cnt = 0.0f, sap = 0.0f, san = 0.0f;
  for (int i = threadIdx.x; i < n; i += 256) {
    float ap = __uint_as_float(apb[i]);
    float an = __uint_as_float(anb[i]);
    loss += fmaxf(ap - an, 0.0f);  // MARGIN = 0
    cnt += (an > ap) ? 1.0f : 0.0f;
    sap += ap;
    san += an;
  }
  sm[0][threadIdx.x] = loss;
  sm[1][threadIdx.x] = cnt;
  sm[2][threadIdx.x] = sap;
  sm[3][threadIdx.x] = san;
  __syncthreads();
  for (int s = 128; s > 0; s >>= 1) {
    if ((int)threadIdx.x < s) {
      sm[0][threadIdx.x] += sm[0][threadIdx.x + s];
      sm[1][threadIdx.x] += sm[1][threadIdx.x + s];
      sm[2][threadIdx.x] += sm[2][threadIdx.x + s];
      sm[3][threadIdx.x] += sm[3][threadIdx.x + s];
    }
    __syncthreads();
  }
  if (threadIdx.x == 0) {
    float inv = 1.0f / (float)n;
    out[0] = sm[0][0] * inv;
    out[1] = sm[1][0] * inv;
    out[2] = sm[2][0] * inv;
    out[3] = sm[3][0] * inv;
  }
}

// ---------------------------------------------------------------------------
extern "C" void kernel_launch(void* const* d_in, const int* in_sizes, int n_in,
                              void* d_out, int out_size, void* d_ws, size_t ws_size,
                              hipStream_t stream) {
  const float* inputs = (const float*)d_in[0];
  const int* targets = (const int*)d_in[1];
  const float* mean_neg = (const float*)d_in[2];
  const int* mean_label = (const int*)d_in[3];

  const int B = in_sizes[1];       // 4096
  const int M = in_sizes[3];       // 1024
  const int n = B + M;             // 5120
  const int ntiles = n / 16;       // 320

  char* ws = (char*)d_ws;
  float* featsW = (float*)ws;      ws += (size_t)n * D_DIM * sizeof(float);
  float* sq = (float*)ws;          ws += (size_t)n * sizeof(float);
  int* lab = (int*)ws;             ws += (size_t)n * sizeof(int);
  unsigned* apb = (unsigned*)ws;   ws += (size_t)n * sizeof(unsigned);
  unsigned* anb = (unsigned*)ws;

  sq_kernel<<<(n + 7) / 8, 256, 0, stream>>>(inputs, mean_neg, sq, B, n);
  prep_labels_kernel<<<(n + 255) / 256, 256, 0, stream>>>(targets, mean_label, lab,
                                                          apb, anb, B, n);
  // ntiles * 16 k-pair waves, 8 waves per block
  swizzle_kernel<<<(ntiles * 16 + 7) / 8, 256, 0, stream>>>(inputs, mean_neg,
                                                            featsW, B, ntiles);
  const int jchunk = (ntiles + NJC - 1) / NJC;     // 16 tiles per chunk
  const int waves = ntiles * NJC;                  // 6400 waves
  gram_mine_kernel<<<(waves + 7) / 8, 256, 0, stream>>>(featsW, sq, lab, apb, anb,
                                                        ntiles, jchunk);
  finalize_kernel<<<1, 256, 0, stream>>>(apb, anb, (float*)d_out, n);
}